// Decoder_32323923870175
// MI455X (gfx1250) — compile-verified
//
#include <hip/hip_runtime.h>
#include <math.h>

// Problem constants (match reference)
#define BB 64
#define TT 512
#define DD 1024
#define VTAG 6
#define VTYPE 20
#define NPAD 32          // padded combined output width (6 tag + 20 type + 6 zero)
#define EE 2048
#define TYPE_PAD_ID 19

typedef float v2f __attribute__((ext_vector_type(2)));
typedef float v8f __attribute__((ext_vector_type(8)));

// ---------------------------------------------------------------------------
// CDNA5 async global->LDS copy (16B per lane), tracked by ASYNCcnt.
// ---------------------------------------------------------------------------
__device__ __forceinline__ void async_copy_b128(const float* gptr, const float* lptr)
{
    const unsigned ldsOff = (unsigned)(unsigned long long)(const void*)lptr;
    asm volatile("global_load_async_to_lds_b128 %0, %1, off"
                 :: "v"(ldsOff), "v"(gptr)
                 : "memory");
}

__device__ __forceinline__ void wait_async_all()
{
    asm volatile("s_wait_asynccnt 0x0" ::: "memory");
}

// ---------------------------------------------------------------------------
// Kernel 1: build combined weight Wc[1024][32] and bias[32]
// cols 0..5 = W_tag, cols 6..25 = W_type, cols 26..31 = 0
// ---------------------------------------------------------------------------
__global__ __launch_bounds__(256) void prep_kernel(
    const float* __restrict__ Wtag, const float* __restrict__ btag,
    const float* __restrict__ Wtype, const float* __restrict__ btype,
    float* __restrict__ Wc, float* __restrict__ bias)
{
    int i = blockIdx.x * blockDim.x + threadIdx.x;
    if (i < DD * NPAD) {
        int k = i >> 5;
        int n = i & 31;
        float v = 0.0f;
        if (n < VTAG)               v = Wtag[k * VTAG + n];
        else if (n < VTAG + VTYPE)  v = Wtype[k * VTYPE + (n - VTAG)];
        Wc[i] = v;
    }
    if (i < NPAD) {
        float bv = 0.0f;
        if (i < VTAG)               bv = btag[i];
        else if (i < VTAG + VTYPE)  bv = btype[i - VTAG];
        bias[i] = bv;
    }
}

// ---------------------------------------------------------------------------
// Kernel 2: fused projection  proj[M=B*T][32] = hidden[M][1024] @ Wc + bias
// via V_WMMA_F32_16X16X4_F32.  One block = 128 rows, 8 waves; K is
// double-buffered through LDS with async global->LDS B128 copies.
// ---------------------------------------------------------------------------
#define TILE_M 128
#define TILE_K 32
#define A_STRIDE 36   // floats; 144B row stride: 16B aligned + conflict-free

__global__ __launch_bounds__(256) void proj_kernel(
    const float* __restrict__ hidden, const float* __restrict__ Wc,
    const float* __restrict__ bias, float* __restrict__ proj)
{
    __shared__ float  ldsA[2][TILE_M][A_STRIDE];      // 2 x 18432 B
    __shared__ float2 ldsB[2][TILE_K / 2][NPAD];      // (K,K+1) pairs, 2 x 4096 B

    const int tid   = threadIdx.x;
    const int lane  = tid & 31;
    const int wave  = tid >> 5;
    const int mHalf = lane >> 4;   // 0: lanes 0-15, 1: lanes 16-31
    const int mIdx  = lane & 15;
    const int rowBase = blockIdx.x * TILE_M;

    // B staging geometry: thread covers pair-row m = tid>>4, cols n, n+1
    const int bM = tid >> 4;
    const int bN = (tid & 15) * 2;

    v8f acc0 = {};  // N = 0..15
    v8f acc1 = {};  // N = 16..31

    // ---- prologue: stage chunk 0 into buffer 0 ----
    {
        #pragma unroll
        for (int j = 0; j < 4; ++j) {
            int i  = tid + j * 256;          // 1024 float4 tiles
            int r  = i >> 3;
            int c4 = (i & 7) * 4;
            async_copy_b128(hidden + (size_t)(rowBase + r) * DD + c4,
                            &ldsA[0][r][c4]);
        }
        const float2 r0 = *(const float2*)(Wc + (size_t)(2 * bM + 0) * NPAD + bN);
        const float2 r1 = *(const float2*)(Wc + (size_t)(2 * bM + 1) * NPAD + bN);
        ldsB[0][bM][bN + 0] = make_float2(r0.x, r1.x);
        ldsB[0][bM][bN + 1] = make_float2(r0.y, r1.y);
    }

    const int nChunks = DD / TILE_K;   // 32
    for (int ci = 0; ci < nChunks; ++ci) {
        const int buf = ci & 1;

        wait_async_all();     // our async copies into `buf` are done
        __syncthreads();      // everyone's copies done + previous compute done

        // ---- issue next chunk's staging (overlaps with WMMA below) ----
        const bool hasNext = (ci + 1) < nChunks;
        float2 nr0, nr1;
        if (hasNext) {
            const int k0n = (ci + 1) * TILE_K;
            #pragma unroll
            for (int j = 0; j < 4; ++j) {
                int i  = tid + j * 256;
                int r  = i >> 3;
                int c4 = (i & 7) * 4;
                async_copy_b128(hidden + (size_t)(rowBase + r) * DD + k0n + c4,
                                &ldsA[buf ^ 1][r][c4]);
            }
            nr0 = *(const float2*)(Wc + (size_t)(k0n + 2 * bM + 0) * NPAD + bN);
            nr1 = *(const float2*)(Wc + (size_t)(k0n + 2 * bM + 1) * NPAD + bN);
        }

        // ---- compute: 8 k-steps x 2 N-tiles = 16 WMMA ----
        const float* arow = &ldsA[buf][wave * 16 + mIdx][0];
        #pragma unroll
        for (int kk8 = 0; kk8 < 8; ++kk8) {
            const int kl = kk8 * 4 + 2 * mHalf;      // even -> 8B aligned
            const int kp = kk8 * 2 + mHalf;          // pair index
            v2f a  = *(const v2f*)(arow + kl);
            v2f b0 = *(const v2f*)&ldsB[buf][kp][mIdx];
            v2f b1 = *(const v2f*)&ldsB[buf][kp][16 + mIdx];
            acc0 = __builtin_amdgcn_wmma_f32_16x16x4_f32(
                false, a, false, b0, (short)0, acc0, false, false);
            acc1 = __builtin_amdgcn_wmma_f32_16x16x4_f32(
                false, a, false, b1, (short)0, acc1, false, false);
        }

        // ---- deferred LDS store of next B chunk (load latency hidden) ----
        if (hasNext) {
            ldsB[buf ^ 1][bM][bN + 0] = make_float2(nr0.x, nr1.x);
            ldsB[buf ^ 1][bM][bN + 1] = make_float2(nr0.y, nr1.y);
        }
    }

    // D layout: VGPR r holds (M = r + 8*mHalf, N = mIdx) per lane
    const int colA = mIdx;
    const int colB = 16 + mIdx;
    const float bA = bias[colA];
    const float bB = bias[colB];
    #pragma unroll
    for (int r = 0; r < 8; ++r) {
        int m = rowBase + wave * 16 + r + 8 * mHalf;
        proj[(size_t)m * NPAD + colA] = acc0[r] + bA;
        proj[(size_t)m * NPAD + colB] = acc1[r] + bB;
    }
}

// ---------------------------------------------------------------------------
// Kernel 3: CRF gold score + forward algorithm; one thread per sample.
// Writes tag_losses to d_out.
// ---------------------------------------------------------------------------
__global__ __launch_bounds__(64) void crf_kernel(
    const float* __restrict__ proj,
    const float* __restrict__ crf_start, const float* __restrict__ crf_end,
    const float* __restrict__ crf_trans,
    const int* __restrict__ tag_ids, const int* __restrict__ seq_lens,
    float* __restrict__ out)
{
    int b = blockIdx.x * blockDim.x + threadIdx.x;
    if (b >= BB) return;

    const float* pb = proj + (size_t)b * TT * NPAD;
    const int*   tg = tag_ids + b * TT;
    const int   len = seq_lens[b];

    // cache the 6x6 transition matrix in registers
    float tr[VTAG][VTAG];
    #pragma unroll
    for (int i = 0; i < VTAG; ++i)
        #pragma unroll
        for (int j = 0; j < VTAG; ++j)
            tr[i][j] = crf_trans[i * VTAG + j];

    // gold path score
    int prev = tg[0];
    float gold = crf_start[prev] + pb[prev];
    for (int t = 1; t < len; ++t) {
        int cur = tg[t];
        gold += pb[t * NPAD + cur] + tr[prev][cur];
        prev = cur;
    }
    gold += crf_end[tg[len - 1]];

    // forward algorithm (log-space)
    float alpha[VTAG], na[VTAG];
    #pragma unroll
    for (int j = 0; j < VTAG; ++j) alpha[j] = crf_start[j] + pb[j];

    for (int t = 1; t < len; ++t) {
        const float* em = pb + t * NPAD;
        #pragma unroll
        for (int j = 0; j < VTAG; ++j) {
            float m = -INFINITY;
            #pragma unroll
            for (int i = 0; i < VTAG; ++i) m = fmaxf(m, alpha[i] + tr[i][j]);
            float s = 0.0f;
            #pragma unroll
            for (int i = 0; i < VTAG; ++i) s += __expf(alpha[i] + tr[i][j] - m);
            na[j] = m + __logf(s) + em[j];
        }
        #pragma unroll
        for (int j = 0; j < VTAG; ++j) alpha[j] = na[j];
    }

    float m = -INFINITY;
    #pragma unroll
    for (int j = 0; j < VTAG; ++j) m = fmaxf(m, alpha[j] + crf_end[j]);
    float s = 0.0f;
    #pragma unroll
    for (int j = 0; j < VTAG; ++j) s += __expf(alpha[j] + crf_end[j] - m);
    float logZ = m + __logf(s);

    out[b] = logZ - gold;
}

// ---------------------------------------------------------------------------
// Kernel 4: per-entity span-mean + log-softmax CE, segment-sum via atomicAdd.
// ---------------------------------------------------------------------------
__global__ __launch_bounds__(256) void ent_kernel(
    const float* __restrict__ proj,
    const int* __restrict__ sid, const int* __restrict__ st,
    const int* __restrict__ en, const int* __restrict__ ty,
    float* __restrict__ out)
{
    int e = blockIdx.x * blockDim.x + threadIdx.x;
    if (e >= EE) return;

    const int t0 = ty[e];
    if (t0 == TYPE_PAD_ID) return;  // valid mask

    const int s = sid[e];
    const int a = st[e];
    const int z = en[e];

    float acc[VTYPE];
    #pragma unroll
    for (int c = 0; c < VTYPE; ++c) acc[c] = 0.0f;

    const float* pb = proj + (size_t)s * TT * NPAD;
    for (int t = a; t < z; ++t) {
        const float* row = pb + t * NPAD + VTAG;  // type cols at offset 6
        #pragma unroll
        for (int c = 0; c < VTYPE; ++c) acc[c] += row[c];
    }
    const float inv = 1.0f / (float)(z - a);
    float m = -INFINITY;
    #pragma unroll
    for (int c = 0; c < VTYPE; ++c) { acc[c] *= inv; m = fmaxf(m, acc[c]); }
    float ssum = 0.0f;
    #pragma unroll
    for (int c = 0; c < VTYPE; ++c) ssum += __expf(acc[c] - m);
    const float ce = (m + __logf(ssum)) - acc[t0];

    atomicAdd(&out[s], ce);
}

// ---------------------------------------------------------------------------
// Launch
// ---------------------------------------------------------------------------
extern "C" void kernel_launch(void* const* d_in, const int* in_sizes, int n_in,
                              void* d_out, int out_size, void* d_ws, size_t ws_size,
                              hipStream_t stream)
{
    const float* full_hidden = (const float*)d_in[0];
    const float* W_tag       = (const float*)d_in[1];
    const float* b_tag       = (const float*)d_in[2];
    const float* W_type      = (const float*)d_in[3];
    const float* b_type      = (const float*)d_in[4];
    const float* crf_start   = (const float*)d_in[5];
    const float* crf_end     = (const float*)d_in[6];
    const float* crf_trans   = (const float*)d_in[7];
    const int*   tag_ids     = (const int*)d_in[8];
    const int*   seq_lens    = (const int*)d_in[9];
    const int*   ent_sid     = (const int*)d_in[10];
    const int*   ent_st      = (const int*)d_in[11];
    const int*   ent_en      = (const int*)d_in[12];
    const int*   ent_ty      = (const int*)d_in[13];
    float*       out         = (float*)d_out;

    // workspace layout (floats)
    float* wsf  = (float*)d_ws;
    float* Wc   = wsf;                         // 1024*32        = 32768
    float* bias = wsf + (size_t)DD * NPAD;     // 32
    float* proj = bias + NPAD;                 // 32768*32       = 1,048,576  (~4.3 MB total)

    // 1) combined weight + bias
    prep_kernel<<<(DD * NPAD + 255) / 256, 256, 0, stream>>>(
        W_tag, b_tag, W_type, b_type, Wc, bias);

    // 2) WMMA projection: 32768 rows / 128 per block = 256 blocks
    proj_kernel<<<(BB * TT) / TILE_M, 256, 0, stream>>>(
        full_hidden, Wc, bias, proj);

    // 3) CRF -> tag_losses into d_out
    crf_kernel<<<1, 64, 0, stream>>>(
        proj, crf_start, crf_end, crf_trans, tag_ids, seq_lens, out);

    // 4) entity CE -> atomicAdd type_losses onto d_out
    ent_kernel<<<(EE + 255) / 256, 256, 0, stream>>>(
        proj, ent_sid, ent_st, ent_en, ent_ty, out);

    (void)in_sizes; (void)n_in; (void)out_size; (void)ws_size;
}